// AngularDescriptor_14869176778801
// MI455X (gfx1250) — compile-verified
//
#include <hip/hip_runtime.h>
#include <math.h>
#include <stdint.h>

// ---------------------------------------------------------------------------
// AngularDescriptor for MI455X (gfx1250, wave32).
//
//  - c_table (4KB) staged to LDS via gfx1250 async global->LDS DMA
//    (global_load_async_to_lds_b128 + s_wait_asynccnt).
//  - Per-lane scalar math: Chebyshev bases, two 8x8 matvecs (LDS float4
//    gathers), Legendre P0..P3.
//  - Outer products h(8) x P(4) for groups of 4 triplets via
//    V_WMMA_F32_16X16X4_F32. Zero padding for A rows 8..15 and B's
//    off-band columns is materialized in LDS so every WMMA operand fetch
//    is an unconditional ds_load_b32 (no EXEC-mask micro-branches).
//  - Segment sum via fp32 global atomics (2.56MB output -> L2 resident).
// ---------------------------------------------------------------------------

#define N_TYPES 4
#define K_MAX   8
#define N_DESC  8
#define L_MAX   4
#define WAVES_PER_BLOCK 8
#define BLOCK_THREADS   256
#define R_C 5.0f
#define HSTRIDE 20   // floats; 80B row stride, 16B aligned, bank-spread

typedef __attribute__((ext_vector_type(2))) float v2f;
typedef __attribute__((ext_vector_type(8))) float v8f;

__device__ __forceinline__ void cheb_basis(float r, float* f) {
    // fc = 0.5*cos(pi*r/rc)+0.5 for r<rc else 0 ; x = 2*(r/rc-1)^2-1
    float fc = (r < R_C) ? (0.5f * cosf(3.14159265358979f * (r / R_C)) + 0.5f)
                         : 0.0f;
    float u = r / R_C - 1.0f;
    float x = 2.0f * u * u - 1.0f;
    float s = 0.5f * fc;
    float tm2 = 1.0f, tm1 = x;
    f[0] = (1.0f + 1.0f) * s;
    f[1] = (x + 1.0f) * s;
#pragma unroll
    for (int k = 2; k < K_MAX; ++k) {
        float tk = 2.0f * x * tm1 - tm2;
        f[k] = (tk + 1.0f) * s;
        tm2 = tm1;
        tm1 = tk;
    }
}

__global__ __launch_bounds__(BLOCK_THREADS)
void angular_descriptor_kernel(const int* __restrict__ trip_idx,   // (n_trip,2)
                               const float* __restrict__ r_ij,
                               const float* __restrict__ r_ik,
                               const float* __restrict__ cth,
                               const int* __restrict__ type_i,
                               const int* __restrict__ type_j,
                               const int* __restrict__ type_k,
                               const float* __restrict__ c_table,  // (4,4,8,8)
                               float* __restrict__ out,            // (n_atoms,8,4)
                               int n_trip)
{
    __shared__ float lds_c[N_TYPES * N_TYPES * N_DESC * K_MAX];     // 4 KB
    __shared__ float s_h16[WAVES_PER_BLOCK][32][HSTRIDE]; // A cols: 16 rows, 8..15 = 0
    __shared__ float s_bp [WAVES_PER_BLOCK][32][HSTRIDE]; // B rows: 16 cols, band = P
    __shared__ int   s_i  [WAVES_PER_BLOCK][32];

    const int tid = threadIdx.x;

    // ---- Stage c_table into LDS with async global->LDS (16B per thread) ----
    {
        // Low 32 bits of the flat shared address == LDS byte offset
        // (ISA 10.2: LDS aperture maps addr[31:0] directly).
        unsigned lds_dst = (unsigned)(uintptr_t)(&lds_c[tid * 4]);
        unsigned long long gsrc = (unsigned long long)(const void*)(c_table + tid * 4);
        asm volatile("global_load_async_to_lds_b128 %0, %1, off"
                     :: "v"(lds_dst), "v"(gsrc) : "memory");
        asm volatile("s_wait_asynccnt 0" ::: "memory");
    }
    __syncthreads();

    const int lane = tid & 31;
    const int w    = tid >> 5;
    const long long tile  = (long long)blockIdx.x * WAVES_PER_BLOCK + w;
    const long long t     = tile * 32 + lane;
    const bool valid = (t < (long long)n_trip);
    const int tc = valid ? (int)t : 0;

    // ---- Per-lane scalar phase -------------------------------------------
    const float rij = r_ij[tc];
    const float rik = r_ik[tc];
    const float c   = cth[tc];
    const int ti = type_i[tc];
    const int tj = type_j[tc];
    const int tk = type_k[tc];
    const int iidx = trip_idx[2 * tc];

    float fij[K_MAX], fik[K_MAX];
    cheb_basis(rij, fij);
    cheb_basis(rik, fik);

    const float4* c4 = (const float4*)lds_c;
    const int pj = (ti * N_TYPES + tj) * 16;  // float4 units: pair*64 floats
    const int pk = (ti * N_TYPES + tk) * 16;

    float h[N_DESC];
#pragma unroll
    for (int d = 0; d < N_DESC; ++d) {
        float4 a0 = c4[pj + 2 * d], a1 = c4[pj + 2 * d + 1];
        float gij = a0.x * fij[0] + a0.y * fij[1] + a0.z * fij[2] + a0.w * fij[3]
                  + a1.x * fij[4] + a1.y * fij[5] + a1.z * fij[6] + a1.w * fij[7];
        float4 b0 = c4[pk + 2 * d], b1 = c4[pk + 2 * d + 1];
        float gik = b0.x * fik[0] + b0.y * fik[1] + b0.z * fik[2] + b0.w * fik[3]
                  + b1.x * fik[4] + b1.y * fik[5] + b1.z * fik[6] + b1.w * fik[7];
        h[d] = (gij * gik);
    }
    // Tail lanes contribute zero through A.
    const float vmask = valid ? 1.0f : 0.0f;
#pragma unroll
    for (int d = 0; d < N_DESC; ++d) h[d] *= vmask;

    // Legendre P0..P3
    const float p0 = 1.0f;
    const float p1 = c;
    const float p2 = (3.0f * c * p1 - p0) * 0.5f;
    const float p3 = (5.0f * c * p2 - 2.0f * p1) * (1.0f / 3.0f);

    // ---- Stage per-wave WMMA operand planes (zeros materialized) ---------
    {
        float4* hrow = (float4*)&s_h16[w][lane][0];
        hrow[0] = make_float4(h[0], h[1], h[2], h[3]);
        hrow[1] = make_float4(h[4], h[5], h[6], h[7]);
        hrow[2] = make_float4(0.f, 0.f, 0.f, 0.f);   // rows 8..11
        hrow[3] = make_float4(0.f, 0.f, 0.f, 0.f);   // rows 12..15

        float4* brow = (float4*)&s_bp[w][lane][0];
        brow[0] = make_float4(0.f, 0.f, 0.f, 0.f);
        brow[1] = make_float4(0.f, 0.f, 0.f, 0.f);
        brow[2] = make_float4(0.f, 0.f, 0.f, 0.f);
        brow[3] = make_float4(0.f, 0.f, 0.f, 0.f);
        brow[lane & 3] = make_float4(p0, p1, p2, p3); // this slot's K-band

        s_i[w][lane] = valid ? iidx : -1;
    }
    __syncthreads();   // ordering fence (all threads reach this; no early exit)

    // Preload segment indices for this lane's column band (branchless).
    const int c_band = (lane & 15) >> 2;
    int idxs[8];
#pragma unroll
    for (int g = 0; g < 8; ++g) idxs[g] = s_i[w][g * 4 + c_band];

    float* const dst0 = out + (lane & 3);
    const int row  = lane & 15;
    const int half = lane >> 4;

    // ---- WMMA outer-product phase: 4 triplets per V_WMMA_F32_16X16X4_F32 --
    // A (16x4 f32, 2 VGPRs): lanes 0-15 hold K=0 (v0) / K=1 (v1),
    //                        lanes 16-31 hold K=2 (v0) / K=3 (v1).
    // B (4x16 f32, 2 VGPRs): mirrored; row k nonzero only in cols [4k,4k+4).
#pragma unroll
    for (int g2 = 0; g2 < 8; ++g2) {
        const int ta = g2 * 4 + half * 2;
        const int k0 = half * 2;
        const int k1 = k0 + 1;
        v2f A, B;
        A.x = s_h16[w][ta    ][row];
        A.y = s_h16[w][ta + 1][row];
        B.x = s_bp [w][g2 * 4 + k0][row];
        B.y = s_bp [w][g2 * 4 + k1][row];

        v8f C = {};
        v8f D = __builtin_amdgcn_wmma_f32_16x16x4_f32(
            /*neg_a=*/false, A, /*neg_b=*/false, B,
            /*c_mod=*/(short)0, C, /*reuse_a=*/false, /*reuse_b=*/false);

        // D: lane n<16 holds column n, rows 0..7 in D[0..7].
        // Column n -> triplet g2*4 + (n>>2), Legendre index n&3.
        // Rows 8..15 (lanes 16-31) are exactly zero -> skipped.
        const int idx = idxs[g2];
        if ((lane < 16) && (idx >= 0)) {
            float* dst = dst0 + (size_t)idx * (N_DESC * L_MAX);
#pragma unroll
            for (int r = 0; r < N_DESC; ++r)
                atomicAdd(dst + r * L_MAX, D[r]);
        }
    }
}

extern "C" void kernel_launch(void* const* d_in, const int* in_sizes, int n_in,
                              void* d_out, int out_size, void* d_ws, size_t ws_size,
                              hipStream_t stream) {
    (void)n_in; (void)d_ws; (void)ws_size;
    // setup_inputs order:
    // 0 n_atoms, 1 triplet_index, 2 r_ij, 3 r_ik, 4 cos_theta,
    // 5 type_i, 6 type_j, 7 type_k, 8 c_table
    const int*   trip_idx = (const int*)  d_in[1];
    const float* r_ij     = (const float*)d_in[2];
    const float* r_ik     = (const float*)d_in[3];
    const float* cth      = (const float*)d_in[4];
    const int*   type_i   = (const int*)  d_in[5];
    const int*   type_j   = (const int*)  d_in[6];
    const int*   type_k   = (const int*)  d_in[7];
    const float* c_table  = (const float*)d_in[8];
    float* out = (float*)d_out;
    const int n_trip = in_sizes[2];

    hipMemsetAsync(d_out, 0, (size_t)out_size * sizeof(float), stream);

    const long long tiles = ((long long)n_trip + 31) / 32;
    const int blocks = (int)((tiles + WAVES_PER_BLOCK - 1) / WAVES_PER_BLOCK);
    angular_descriptor_kernel<<<blocks, BLOCK_THREADS, 0, stream>>>(
        trip_idx, r_ij, r_ik, cth, type_i, type_j, type_k, c_table, out, n_trip);
}